// ResidualEdgeGATConv_15058155340554
// MI455X (gfx1250) — compile-verified
//
#include <hip/hip_runtime.h>
#include <math.h>

// ---------------- problem constants (match reference) ----------------
#define N_NODES   10000
#define N_EDGES   640000
#define IN_FEATS  128
#define OUT_FEATS 128
#define NUM_HEADS 4
#define D_HEAD    32
#define NEG_SLOPE 0.2f

typedef float v2f __attribute__((ext_vector_type(2)));
typedef float v8f __attribute__((ext_vector_type(8)));

// Exact float atomic max via int/uint atomics (monotone bit-pattern trick).
__device__ __forceinline__ void atomicMaxF(float* addr, float val) {
    if (val >= 0.0f) atomicMax((int*)addr, __float_as_int(val));
    else             atomicMin((unsigned int*)addr, (unsigned int)__float_as_int(val));
}

// ---------------- kernel 1: per-call state init ----------------
__global__ void k_init(float* __restrict__ emax, float* __restrict__ denom) {
    int i = blockIdx.x * blockDim.x + threadIdx.x;
    if (i < N_NODES * NUM_HEADS) {
        emax[i]  = -1e30f;
        denom[i] = 0.0f;
    }
}

// ---------------- kernel 2: fused projection GEMM (WMMA f32 16x16x4) -------
// One wave computes one 16x16 tile of [h | res] = feat @ [W | W_res].
// N_NODES = 625*16 exactly -> no M remainder. 256 output cols -> 16 N-tiles.
// A-frag layout (16x4 f32): lane<16 -> K = k0+{0,1}; lane>=16 -> K = k0+{2,3}; M = lane&15.
// B-frag layout (4x16 f32): same K split; N = lane&15.
// C layout: vgpr i, lane<16 -> M=i; lane>=16 -> M=i+8; N = lane&15.
__global__ void k_gemm(const float* __restrict__ feat,
                       const float* __restrict__ W,
                       const float* __restrict__ Wres,
                       const float* __restrict__ bias,
                       float* __restrict__ h,      // N x 128 scratch
                       float* __restrict__ out)    // N x 128 (residual + bias)
{
    const int wave = blockIdx.x * 4 + (threadIdx.x >> 5);  // 4 waves / block
    const int lane = threadIdx.x & 31;
    const int tileM = wave >> 4;       // 0..624
    const int tileN = wave & 15;       // 0..15 over 256 fused cols
    const int m0 = tileM * 16;
    const int n0 = tileN * 16;

    const float* B   = (n0 < OUT_FEATS) ? W : Wres;        // wave-uniform
    const int bcol0  = (n0 < OUT_FEATS) ? n0 : (n0 - OUT_FEATS);

    const int mrow = m0 + (lane & 15);
    const int half = lane >> 4;        // 0 or 1 (K split)
    const int col  = bcol0 + (lane & 15);

    v8f c = {};
#pragma unroll
    for (int k0 = 0; k0 < IN_FEATS; k0 += 4) {
        const int k = k0 + 2 * half;
        v2f a = *(const v2f*)(feat + mrow * IN_FEATS + k);   // adjacent K pair
        v2f b;
        b.x = B[(k    ) * OUT_FEATS + col];
        b.y = B[(k + 1) * OUT_FEATS + col];
        c = __builtin_amdgcn_wmma_f32_16x16x4_f32(
                /*neg_a=*/false, a, /*neg_b=*/false, b,
                /*c_mod=*/(short)0, c, /*reuse_a=*/false, /*reuse_b=*/false);
    }

    if (n0 < OUT_FEATS) {
#pragma unroll
        for (int i = 0; i < 8; ++i) {
            const int r = m0 + i + 8 * half;
            h[r * OUT_FEATS + col] = c[i];
        }
    } else {
        const float bv = bias[col];
#pragma unroll
        for (int i = 0; i < 8; ++i) {
            const int r = m0 + i + 8 * half;
            out[r * OUT_FEATS + col] = c[i] + bv;
        }
    }
}

// ---------------- kernel 3: per-node attention logits ----------------
// One wave per node; lane = d within a head; shuffle-reduce over 32 lanes.
__global__ void k_logits(const float* __restrict__ h,
                         const float* __restrict__ attn_l,
                         const float* __restrict__ attn_r,
                         float* __restrict__ el, float* __restrict__ er)
{
    const int node = blockIdx.x * (blockDim.x >> 5) + (threadIdx.x >> 5);
    const int lane = threadIdx.x & 31;
    if (node >= N_NODES) return;
#pragma unroll
    for (int hd = 0; hd < NUM_HEADS; ++hd) {
        const float v  = h[node * OUT_FEATS + hd * D_HEAD + lane];
        float pl = v * attn_l[hd * D_HEAD + lane];
        float pr = v * attn_r[hd * D_HEAD + lane];
#pragma unroll
        for (int off = 16; off > 0; off >>= 1) {
            pl += __shfl_xor(pl, off, 32);
            pr += __shfl_xor(pr, off, 32);
        }
        if (lane == 0) {
            el[node * NUM_HEADS + hd] = pl;
            er[node * NUM_HEADS + hd] = pr;
        }
    }
}

// ---------------- kernel 4: segmented max over edges ----------------
__global__ void k_edge_max(const int* __restrict__ src, const int* __restrict__ dst,
                           const float* __restrict__ el, const float* __restrict__ er,
                           float* __restrict__ emax)
{
    const int e = blockIdx.x * blockDim.x + threadIdx.x;
    if (e >= N_EDGES) return;
    const int s = src[e], d = dst[e];
    const float4 l4 = *(const float4*)(el + s * NUM_HEADS);
    const float4 r4 = *(const float4*)(er + d * NUM_HEADS);
    float ev[4] = { l4.x + r4.x, l4.y + r4.y, l4.z + r4.z, l4.w + r4.w };
#pragma unroll
    for (int hd = 0; hd < NUM_HEADS; ++hd) {
        float x = ev[hd];
        x = (x > 0.0f) ? x : NEG_SLOPE * x;
        atomicMaxF(emax + d * NUM_HEADS + hd, x);
    }
}

// ---------------- kernel 5: exp + segmented sum ----------------
__global__ void k_edge_expsum(const int* __restrict__ src, const int* __restrict__ dst,
                              const float* __restrict__ el, const float* __restrict__ er,
                              const float* __restrict__ emax, float* __restrict__ denom)
{
    const int e = blockIdx.x * blockDim.x + threadIdx.x;
    if (e >= N_EDGES) return;
    const int s = src[e], d = dst[e];
    const float4 l4 = *(const float4*)(el + s * NUM_HEADS);
    const float4 r4 = *(const float4*)(er + d * NUM_HEADS);
    float ev[4] = { l4.x + r4.x, l4.y + r4.y, l4.z + r4.z, l4.w + r4.w };
#pragma unroll
    for (int hd = 0; hd < NUM_HEADS; ++hd) {
        float x = ev[hd];
        x = (x > 0.0f) ? x : NEG_SLOPE * x;
        const float w = __expf(x - emax[d * NUM_HEADS + hd]);
        atomicAdd(denom + d * NUM_HEADS + hd, w);
    }
}

// ---------------- kernel 6: weighted scatter of messages ----------------
// One block per edge, one wave per head; lane = d. src/dst indexed by
// blockIdx -> scalar loads; per-head logits are wave-uniform scalar loads.
__global__ void k_scatter(const int* __restrict__ src, const int* __restrict__ dst,
                          const float* __restrict__ el, const float* __restrict__ er,
                          const float* __restrict__ emax, const float* __restrict__ denom,
                          const float* __restrict__ h, float* __restrict__ out)
{
    const int e    = blockIdx.x;
    const int hd   = threadIdx.x >> 5;
    const int lane = threadIdx.x & 31;
    const int s = src[e], d = dst[e];
    float x = el[s * NUM_HEADS + hd] + er[d * NUM_HEADS + hd];
    x = (x > 0.0f) ? x : NEG_SLOPE * x;
    const float a = __expf(x - emax[d * NUM_HEADS + hd]) / denom[d * NUM_HEADS + hd];
    const float v = a * h[s * OUT_FEATS + hd * D_HEAD + lane];
    atomicAdd(out + d * OUT_FEATS + hd * D_HEAD + lane, v);
}

// ---------------- launcher ----------------
extern "C" void kernel_launch(void* const* d_in, const int* in_sizes, int n_in,
                              void* d_out, int out_size, void* d_ws, size_t ws_size,
                              hipStream_t stream) {
    (void)in_sizes; (void)n_in; (void)out_size; (void)ws_size;
    const float* feat   = (const float*)d_in[0];
    const int*   src    = (const int*)  d_in[1];
    const int*   dst    = (const int*)  d_in[2];
    const float* W      = (const float*)d_in[3];
    const float* attn_l = (const float*)d_in[4];
    const float* attn_r = (const float*)d_in[5];
    const float* Wres   = (const float*)d_in[6];
    const float* bias   = (const float*)d_in[7];
    float* out = (float*)d_out;

    // workspace layout (floats): h[N*128] | el[N*4] | er[N*4] | emax[N*4] | denom[N*4]
    float* ws    = (float*)d_ws;
    float* h     = ws;
    float* el    = h    + (size_t)N_NODES * OUT_FEATS;
    float* er    = el   + (size_t)N_NODES * NUM_HEADS;
    float* emax  = er   + (size_t)N_NODES * NUM_HEADS;
    float* denom = emax + (size_t)N_NODES * NUM_HEADS;

    k_init<<<(N_NODES * NUM_HEADS + 255) / 256, 256, 0, stream>>>(emax, denom);

    // 625 M-tiles * 16 N-tiles = 10000 waves; 4 waves/block
    k_gemm<<<(N_NODES / 16) * 16 / 4, 128, 0, stream>>>(feat, W, Wres, bias, h, out);

    // one wave per node, 4 waves/block
    k_logits<<<(N_NODES + 3) / 4, 128, 0, stream>>>(h, attn_l, attn_r, el, er);

    k_edge_max<<<(N_EDGES + 255) / 256, 256, 0, stream>>>(src, dst, el, er, emax);
    k_edge_expsum<<<(N_EDGES + 255) / 256, 256, 0, stream>>>(src, dst, el, er, emax, denom);

    // one block per edge, one wave per head
    k_scatter<<<N_EDGES, NUM_HEADS * 32, 0, stream>>>(src, dst, el, er, emax, denom, h, out);
}